// MambaBlock_51969104281934
// MI455X (gfx1250) — compile-verified
//
#include <hip/hip_runtime.h>
#include <hip/hip_bf16.h>

// ---------------------------------------------------------------------------
// Mamba block for MI455X (gfx1250, wave32, WMMA + Tensor Data Mover).
// GEMMs: V_WMMA_F32_16X16X4_F32, 64x64 block tiles, double-buffered LDS
//        panels filled by TENSOR_LOAD_TO_LDS (TDM) with pad_interval=32dw /
//        pad_amount=1dw giving the bank-conflict-free stride-33 layout.
// Scan:  4 lanes/channel, v_exp_f32, shfl_xor butterfly reduction.
// ---------------------------------------------------------------------------

typedef __attribute__((ext_vector_type(2))) float v2f;
typedef __attribute__((ext_vector_type(8))) float v8f;
typedef __attribute__((ext_vector_type(4))) unsigned int u32x4;
typedef __attribute__((ext_vector_type(4))) int i32x4;
typedef __attribute__((ext_vector_type(8))) int i32x8;

#define SEQ_LEN 2048
#define N_EMBD  2048
#define D_INNER 4096
#define DT_RANK 128
#define D_STATE 16
#define D_CONV  4
#define DBC_W   (DT_RANK + 2 * D_STATE)   // 160

#define TM 64
#define TN 64
#define TKC 32

#if defined(__gfx1250__) && __has_builtin(__builtin_amdgcn_tensor_load_to_lds)
#define USE_TDM 1
#if __has_include(<hip/amd_detail/amd_gfx1250_TDM.h>)
#define TDM_SIX 1
#else
#define TDM_SIX 0
#endif
#else
#define USE_TDM 0
#endif

__device__ __forceinline__ float silu_f(float x) {
    return x / (1.0f + expf(-x));
}

#if USE_TDM
// Issue one TDM descriptor: copy a (64 rows x 32 cols) fp32 tile whose start
// is gsrc (row stride = stride_elems floats) into LDS at lds_byte, writing
// rows with stride 33 dwords (pad 1 dword after every 32). Rows/cols beyond
// (rows_rem, cols_rem) read as zero (TDM OOB semantics).
__device__ __forceinline__ void tdm_load_tile(const float* gsrc, unsigned lds_byte,
                                              unsigned rows_rem, unsigned cols_rem,
                                              unsigned stride_elems) {
    unsigned long long ga = (unsigned long long)(size_t)gsrc;
    u32x4 g0;
    g0.x = 1u;                                            // count=1, user desc
    g0.y = lds_byte;                                      // lds_addr (bytes)
    g0.z = (unsigned)(ga & 0xFFFFFFFFu);                  // global_addr[31:0]
    g0.w = (unsigned)((ga >> 32) & 0x01FFFFFFu) | (2u << 30); // addr[56:32] | type=2
    unsigned long long st = (unsigned long long)stride_elems;
    i32x8 g1;
    g1[0] = (int)((2u << 16)          // data_size = 4 bytes
                | (1u << 20)          // pad_enable
                | (4u << 22));        // pad_interval: 32 dwords
                                      // pad_amount encoded 0 -> 1 dword
    g1[1] = (int)((cols_rem & 0xFFFFu) << 16);                    // tensor_dim0[15:0]
    g1[2] = (int)((cols_rem >> 16) | ((rows_rem & 0xFFFFu) << 16)); // dim0 hi | dim1 lo
    g1[3] = (int)((rows_rem >> 16) | ((unsigned)TKC << 16));      // dim1 hi | tile_dim0
    g1[4] = (int)((unsigned)TM);                                  // tile_dim1 | tile_dim2=0
    g1[5] = (int)(st & 0xFFFFFFFFu);                              // dim0_stride lo
    g1[6] = (int)((st >> 32) & 0xFFFFu);                          // dim0_stride hi
    g1[7] = 0;
    i32x4 z4 = {0, 0, 0, 0};
#if TDM_SIX
    i32x8 z8 = {0, 0, 0, 0, 0, 0, 0, 0};
    __builtin_amdgcn_tensor_load_to_lds(g0, g1, z4, z4, z8, 0);
#else
    __builtin_amdgcn_tensor_load_to_lds(g0, g1, z4, z4, 0);
#endif
}
#endif // USE_TDM

// C[m][n] = epilogue( sum_k A[m][k] * W[n][k] + bias[n] )
// A: M x K (row stride lda), W: N x K (row stride K), C: M x N (row stride ldc)
// epilogue: 0 = identity, 1 = softplus.  K must be a multiple of 32.
__global__ __launch_bounds__(128)
void gemm_xwt_wmma(const float* __restrict__ A, int lda,
                   const float* __restrict__ W,
                   const float* __restrict__ bias,
                   float* __restrict__ C, int ldc,
                   int M, int N, int K, int epilogue) {
    __shared__ float sA[2][TM][TKC + 1];   // stride 33: no LDS bank conflicts
    __shared__ float sW[2][TN][TKC + 1];

    const int tid  = threadIdx.x;
    const int wid  = tid >> 5;          // wave 0..3 -> rows w*16..w*16+15
    const int lane = tid & 31;
    const int hl   = lane & 15;
    const int kh   = lane >> 4;         // 0: K {0,1}/M 0..7, 1: K {2,3}/M 8..15

    const int row0 = blockIdx.y * TM;
    const int col0 = blockIdx.x * TN;
    const int nchunks = K / TKC;

    v8f acc0 = 0.0f, acc1 = 0.0f, acc2 = 0.0f, acc3 = 0.0f;

#if USE_TDM
    const unsigned ldsA0 = (unsigned)(size_t)&sA[0][0][0];
    const unsigned ldsA1 = (unsigned)(size_t)&sA[1][0][0];
    const unsigned ldsW0 = (unsigned)(size_t)&sW[0][0][0];
    const unsigned ldsW1 = (unsigned)(size_t)&sW[1][0][0];
    if (tid < 32) {   // wave 0 drives the Tensor Data Mover
        tdm_load_tile(&A[(size_t)row0 * lda], ldsA0,
                      (unsigned)(M - row0), (unsigned)K, (unsigned)lda);
        tdm_load_tile(&W[(size_t)col0 * K], ldsW0,
                      (unsigned)(N > col0 ? N - col0 : 0), (unsigned)K, (unsigned)K);
    }
#endif

    for (int c = 0; c < nchunks; ++c) {
        const int k0  = c * TKC;
        const int buf = c & 1;

#if USE_TDM
        if (tid < 32) {
            if (c + 1 < nchunks) {
                const int k1 = k0 + TKC;
                tdm_load_tile(&A[(size_t)row0 * lda + k1], buf ? ldsA0 : ldsA1,
                              (unsigned)(M - row0), (unsigned)(K - k1), (unsigned)lda);
                tdm_load_tile(&W[(size_t)col0 * K + k1], buf ? ldsW0 : ldsW1,
                              (unsigned)(N > col0 ? N - col0 : 0),
                              (unsigned)(K - k1), (unsigned)K);
                __builtin_amdgcn_s_wait_tensorcnt(2);  // chunk c's 2 DMAs done
            } else {
                __builtin_amdgcn_s_wait_tensorcnt(0);
            }
        }
#else
        // Synchronous fallback: 128 threads stage both 64x32 panels (float4).
#pragma unroll
        for (int i = 0; i < 4; ++i) {
            int g  = tid + i * 128;
            int r  = g >> 3;
            int cg = (g & 7) << 2;
            int gm = row0 + r;
            float4 av = make_float4(0.f, 0.f, 0.f, 0.f);
            if (gm < M)
                av = *reinterpret_cast<const float4*>(&A[(size_t)gm * lda + k0 + cg]);
            sA[buf][r][cg + 0] = av.x; sA[buf][r][cg + 1] = av.y;
            sA[buf][r][cg + 2] = av.z; sA[buf][r][cg + 3] = av.w;
            int gn = col0 + r;
            float4 wv = make_float4(0.f, 0.f, 0.f, 0.f);
            if (gn < N)
                wv = *reinterpret_cast<const float4*>(&W[(size_t)gn * K + k0 + cg]);
            sW[buf][r][cg + 0] = wv.x; sW[buf][r][cg + 1] = wv.y;
            sW[buf][r][cg + 2] = wv.z; sW[buf][r][cg + 3] = wv.w;
        }
#endif
        __syncthreads();

        const int mrow = wid * 16 + hl;
#pragma unroll
        for (int kk = 0; kk < TKC; kk += 4) {
            const int kb = kk + 2 * kh;
            v2f a; a[0] = sA[buf][mrow][kb]; a[1] = sA[buf][mrow][kb + 1];

            v2f b0; b0[0] = sW[buf][hl +  0][kb]; b0[1] = sW[buf][hl +  0][kb + 1];
            v2f b1; b1[0] = sW[buf][hl + 16][kb]; b1[1] = sW[buf][hl + 16][kb + 1];
            v2f b2; b2[0] = sW[buf][hl + 32][kb]; b2[1] = sW[buf][hl + 32][kb + 1];
            v2f b3; b3[0] = sW[buf][hl + 48][kb]; b3[1] = sW[buf][hl + 48][kb + 1];

            acc0 = __builtin_amdgcn_wmma_f32_16x16x4_f32(false, a, false, b0, (short)0, acc0, false, false);
            acc1 = __builtin_amdgcn_wmma_f32_16x16x4_f32(false, a, false, b1, (short)0, acc1, false, false);
            acc2 = __builtin_amdgcn_wmma_f32_16x16x4_f32(false, a, false, b2, (short)0, acc2, false, false);
            acc3 = __builtin_amdgcn_wmma_f32_16x16x4_f32(false, a, false, b3, (short)0, acc3, false, false);
        }
        __syncthreads();
    }

    // epilogue + store: element v of a tile lives at (m = v + 8*kh, n = hl)
    v8f accs[4] = {acc0, acc1, acc2, acc3};
#pragma unroll
    for (int t = 0; t < 4; ++t) {
        int gn = col0 + t * 16 + hl;
        if (gn >= N) continue;
        float bv = bias ? bias[gn] : 0.0f;
#pragma unroll
        for (int v = 0; v < 8; ++v) {
            int gm = row0 + wid * 16 + 8 * kh + v;
            if (gm >= M) continue;
            float val = accs[t][v] + bv;
            if (epilogue == 1)                 // softplus
                val = (val > 20.0f) ? val : log1pf(expf(val));
            C[(size_t)gm * ldc + gn] = val;
        }
    }
}

// xc[l][c] = silu( conv_b[c] + sum_j conv_w[c][j] * xi[l-3+j][c] ), xi = xr[:, :4096]
__global__ __launch_bounds__(256)
void conv_silu_kernel(const float* __restrict__ xr,
                      const float* __restrict__ cw,
                      const float* __restrict__ cb,
                      float* __restrict__ xc) {
    int idx = blockIdx.x * 256 + threadIdx.x;
    if (idx >= SEQ_LEN * D_INNER) return;
    int l = idx / D_INNER, ch = idx % D_INNER;
    float acc = cb[ch];
#pragma unroll
    for (int j = 0; j < D_CONV; ++j) {
        int ll = l - (D_CONV - 1) + j;
        if (ll >= 0)
            acc += cw[ch * D_CONV + j] * xr[(size_t)ll * (2 * D_INNER) + ch];
    }
    xc[idx] = silu_f(acc);
}

// y[l][c] *= silu(res[l][c]), res = xr[:, 4096:]
__global__ __launch_bounds__(256)
void gate_kernel(float* __restrict__ y, const float* __restrict__ xr) {
    int idx = blockIdx.x * 256 + threadIdx.x;
    if (idx >= SEQ_LEN * D_INNER) return;
    int l = idx / D_INNER, ch = idx % D_INNER;
    float r = xr[(size_t)l * (2 * D_INNER) + D_INNER + ch];
    y[idx] *= silu_f(r);
}

// Selective scan: 4 lanes per channel, 4 states per lane, butterfly reduce.
__global__ __launch_bounds__(256)
void scan_kernel(const float* __restrict__ u,      // xc  (L x di)
                 const float* __restrict__ delta,  // (L x di)
                 const float* __restrict__ dbc,    // (L x 160); B at +128, C at +144
                 const float* __restrict__ A_log,  // (di x 16)
                 const float* __restrict__ Dv,     // (di)
                 float* __restrict__ y) {          // (L x di)
    int idx = blockIdx.x * 256 + threadIdx.x;      // 16384 threads
    int ch  = idx >> 2;
    int sub = idx & 3;                             // states s = 4*sub .. 4*sub+3
    if (ch >= D_INNER) return;

    float Arow[4], h[4];
#pragma unroll
    for (int s = 0; s < 4; ++s) {
        Arow[s] = -__expf(A_log[ch * D_STATE + 4 * sub + s]);
        h[s] = 0.0f;
    }
    const float Dch = Dv[ch];

    for (int l = 0; l < SEQ_LEN; ++l) {
        const float d  = delta[(size_t)l * D_INNER + ch];
        const float uu = u[(size_t)l * D_INNER + ch];
        const float du = d * uu;
        const float* Br = dbc + (size_t)l * DBC_W + DT_RANK;
        const float* Cr = Br + D_STATE;
        float acc = 0.0f;
#pragma unroll
        for (int s = 0; s < 4; ++s) {
            h[s] = __expf(d * Arow[s]) * h[s] + du * Br[4 * sub + s];
            acc += h[s] * Cr[4 * sub + s];
        }
        acc += __shfl_xor(acc, 1, 32);
        acc += __shfl_xor(acc, 2, 32);
        if (sub == 0)
            y[(size_t)l * D_INNER + ch] = acc + uu * Dch;
    }
}

extern "C" void kernel_launch(void* const* d_in, const int* in_sizes, int n_in,
                              void* d_out, int out_size, void* d_ws, size_t ws_size,
                              hipStream_t stream) {
    const float* x         = (const float*)d_in[0];
    const float* in_proj_w = (const float*)d_in[1];
    const float* in_proj_b = (const float*)d_in[2];
    const float* conv_w    = (const float*)d_in[3];
    const float* conv_b    = (const float*)d_in[4];
    const float* ssm_in_w  = (const float*)d_in[5];
    const float* delta_w   = (const float*)d_in[6];
    const float* A_log     = (const float*)d_in[7];
    const float* Dvec      = (const float*)d_in[8];
    const float* out_w     = (const float*)d_in[9];
    const float* out_b     = (const float*)d_in[10];
    float* out = (float*)d_out;

    float* ws = (float*)d_ws;
    float* xr    = ws;                                    // L x 8192
    float* xc    = xr    + (size_t)SEQ_LEN * 2 * D_INNER; // L x 4096
    float* dbc   = xc    + (size_t)SEQ_LEN * D_INNER;     // L x 160
    float* delta = dbc   + (size_t)SEQ_LEN * DBC_W;       // L x 4096
    float* y     = delta + (size_t)SEQ_LEN * D_INNER;     // L x 4096

    dim3 blk(128);
    // 1) xr = x @ in_proj_w.T + in_proj_b
    gemm_xwt_wmma<<<dim3((2 * D_INNER) / TN, SEQ_LEN / TM), blk, 0, stream>>>(
        x, N_EMBD, in_proj_w, in_proj_b, xr, 2 * D_INNER,
        SEQ_LEN, 2 * D_INNER, N_EMBD, 0);

    // 2) depthwise causal conv + SiLU
    int ew_blocks = (SEQ_LEN * D_INNER + 255) / 256;
    conv_silu_kernel<<<ew_blocks, 256, 0, stream>>>(xr, conv_w, conv_b, xc);

    // 3) dbc = xc @ ssm_in_w.T
    gemm_xwt_wmma<<<dim3((DBC_W + TN - 1) / TN, SEQ_LEN / TM), blk, 0, stream>>>(
        xc, D_INNER, ssm_in_w, nullptr, dbc, DBC_W,
        SEQ_LEN, DBC_W, D_INNER, 0);

    // 4) delta = softplus(dbc[:, :128] @ delta_w.T)
    gemm_xwt_wmma<<<dim3(D_INNER / TN, SEQ_LEN / TM), blk, 0, stream>>>(
        dbc, DBC_W, delta_w, nullptr, delta, D_INNER,
        SEQ_LEN, D_INNER, DT_RANK, 1);

    // 5) selective scan
    scan_kernel<<<(D_INNER * 4) / 256, 256, 0, stream>>>(
        xc, delta, dbc, A_log, Dvec, y);

    // 6) y *= silu(res)
    gate_kernel<<<ew_blocks, 256, 0, stream>>>(y, xr);

    // 7) out = y @ out_w.T + out_b
    gemm_xwt_wmma<<<dim3(N_EMBD / TN, SEQ_LEN / TM), blk, 0, stream>>>(
        y, D_INNER, out_w, out_b, out, N_EMBD,
        SEQ_LEN, N_EMBD, D_INNER, 0);
}